// AxialSelfAttention2d_48893907698062
// MI455X (gfx1250) — compile-verified
//
#include <hip/hip_runtime.h>
#include <hip/hip_bf16.h>

typedef float v2f __attribute__((ext_vector_type(2)));
typedef float v8f __attribute__((ext_vector_type(8)));

constexpr int H  = 4;
constexpr int C  = 32;
constexpr int D  = 128;   // H*C
constexpr int S  = 128;
constexpr int L  = 256;
constexpr int SL = S * L; // 32768

// D = A*B + C, A:16x4 f32, B:4x16 f32, C/D:16x16 f32  (CDNA5 V_WMMA_F32_16X16X4_F32)
__device__ __forceinline__ v8f wmma4(v2f a, v2f b, v8f c) {
  return __builtin_amdgcn_wmma_f32_16x16x4_f32(false, a, false, b, (short)0, c,
                                               false, false);
}

// ---- CDNA5 async global->LDS copies (ASYNCcnt-tracked), per CDNA5 ISA ----
__device__ __forceinline__ unsigned lds_off(const void* p) {
  // generic LDS pointer = {SHARED_BASE hi, lds_offset[31:0]}; low 32 bits are the DS offset
  return (unsigned)(unsigned long long)p;
}
__device__ __forceinline__ void async_g2l_b32(unsigned lds, const float* g) {
  asm volatile("global_load_async_to_lds_b32 %0, %1, off"
               :: "v"(lds), "v"(g) : "memory");
}
__device__ __forceinline__ void async_g2l_b128(unsigned lds, const float* g) {
  asm volatile("global_load_async_to_lds_b128 %0, %1, off"
               :: "v"(lds), "v"(g) : "memory");
}
__device__ __forceinline__ void async_wait0() {
  asm volatile("s_wait_asynccnt 0" ::: "memory");
}

// ---------------------------------------------------------------------------
// QKV projection: out[o, n] = bias[o] + sum_d w[o*128+d] * x[d*SL + n]
// M=384 (3 blocks of 128 via blockIdx.y), N=SL (64 per block), K=128.
// ---------------------------------------------------------------------------
__global__ __launch_bounds__(256)
void qkv_gemm_kernel(const float* __restrict__ x, const float* __restrict__ w,
                     const float* __restrict__ bias, float* __restrict__ out) {
  // pad 132: row stride 528 B (16B-aligned for b128 async) and A-frag reads
  // advance 4 banks per lane -> 16 lanes hit 16 distinct banks.
  __shared__ float Wlds[128][132];
  __shared__ float Xlds[128][64];
  __shared__ float Blds[128];

  const int tid  = threadIdx.x;
  const int lane = tid & 31;
  const int wave = tid >> 5;
  const int half = lane >> 4;
  const int lm   = lane & 15;

  const int n0 = blockIdx.x * 64;
  const int m0 = blockIdx.y * 128;

  // async-stage W tile (128x128) in 16B chunks
  for (int e4 = tid; e4 < 128 * 32; e4 += 256) {
    int m = e4 >> 5, k = (e4 & 31) * 4;
    async_g2l_b128(lds_off(&Wlds[m][k]), &w[(m0 + m) * 128 + k]);
  }
  // async-stage X tile (128x64) in 16B chunks
  for (int e4 = tid; e4 < 128 * 16; e4 += 256) {
    int k = e4 >> 4, nn = (e4 & 15) * 4;
    async_g2l_b128(lds_off(&Xlds[k][nn]), &x[k * SL + n0 + nn]);
  }
  if (tid < 128) Blds[tid] = bias[m0 + tid];
  async_wait0();
  __syncthreads();

  const int mt = wave;  // 8 waves -> 8 M-tiles of 16
  for (int nt = 0; nt < 4; ++nt) {
    v8f acc = {};
#pragma unroll 4
    for (int kk = 0; kk < 32; ++kk) {
      const int kb = kk * 4 + half * 2;
      v2f a, b;
      a.x = Wlds[mt * 16 + lm][kb];
      a.y = Wlds[mt * 16 + lm][kb + 1];
      b.x = Xlds[kb][nt * 16 + lm];
      b.y = Xlds[kb + 1][nt * 16 + lm];
      acc = wmma4(a, b, acc);
    }
#pragma unroll
    for (int r = 0; r < 8; ++r) {
      const int m = mt * 16 + r + half * 8;
      out[(m0 + m) * SL + n0 + nt * 16 + lm] = acc[r] + Blds[m];
    }
  }
}

// ---------------------------------------------------------------------------
// Axial attention along one axis of length LEN (256 for rows, 128 for cols).
// One workgroup = (h, other, 64-wide i-block). Element (c,pos) lives at
//   qkv[sec*D*SL + (h*C+c)*SL + other*otherStride + pos*posStride]
// scores[i,j] = sum_c q[c,i] k[c,j];  softmax over j;  out[c,i]=sum_j a[i,j]v[c,j]
// ---------------------------------------------------------------------------
template <int LEN>
__global__ __launch_bounds__(256)
void attn_kernel(const float* __restrict__ qkv, float* __restrict__ out,
                 int posStride, int otherStride, int otherLen, int iblocks) {
  __shared__ float Klds[C][LEN];
  __shared__ float Vlds[C][LEN + 1];  // pad: AV A-frag reads go down rows
  __shared__ float Qlds[C][64];
  __shared__ float Sc[64][LEN + 1];   // pad: AV B-frag reads go down rows

  const int tid  = threadIdx.x;
  const int lane = tid & 31;
  const int wave = tid >> 5;
  const int half = lane >> 4;
  const int lm   = lane & 15;

  const int bid   = blockIdx.x;
  const int iblk  = bid % iblocks;
  const int other = (bid / iblocks) % otherLen;
  const int h     = bid / (iblocks * otherLen);
  const int i0    = iblk * 64;

  const int qoff = h * C * SL + other * otherStride;
  const int koff = qoff + D * SL;
  const int voff = qoff + 2 * D * SL;

  // ---- async-stage Q (32x64), K (32xLEN), V (32xLEN) directly into LDS ----
  for (int e = tid; e < C * 64; e += 256) {
    int c = e >> 6, ii = e & 63;
    async_g2l_b32(lds_off(&Qlds[c][ii]),
                  &qkv[qoff + c * SL + (i0 + ii) * posStride]);
  }
  for (int e = tid; e < C * LEN; e += 256) {
    int c = e / LEN, j = e % LEN;
    async_g2l_b32(lds_off(&Klds[c][j]), &qkv[koff + c * SL + j * posStride]);
  }
  for (int e = tid; e < C * LEN; e += 256) {
    int c = e / LEN, j = e % LEN;
    async_g2l_b32(lds_off(&Vlds[c][j]), &qkv[voff + c * SL + j * posStride]);
  }
  async_wait0();
  __syncthreads();

  // ---- scores = Q^T K  (tiles 16x16, K-dim = 32 -> 8 WMMAs) ----
  constexpr int JT = LEN / 16;
  for (int t = wave; t < 4 * JT; t += 8) {
    const int it = t / JT, jt = t % JT;
    v8f acc = {};
#pragma unroll
    for (int kk = 0; kk < 8; ++kk) {
      const int cb = kk * 4 + half * 2;
      v2f a, b;
      a.x = Qlds[cb][it * 16 + lm];
      a.y = Qlds[cb + 1][it * 16 + lm];
      b.x = Klds[cb][jt * 16 + lm];
      b.y = Klds[cb + 1][jt * 16 + lm];
      acc = wmma4(a, b, acc);
    }
#pragma unroll
    for (int r = 0; r < 8; ++r)
      Sc[it * 16 + r + half * 8][jt * 16 + lm] = acc[r];
  }
  __syncthreads();

  // ---- softmax over j (one thread per row; rows hit distinct banks) ----
  if (tid < 64) {
    float m = -3.4e38f;
    for (int j = 0; j < LEN; ++j) m = fmaxf(m, Sc[tid][j]);
    float s = 0.f;
    for (int j = 0; j < LEN; ++j) {
      float e = expf(Sc[tid][j] - m);
      Sc[tid][j] = e;
      s += e;
    }
    const float inv = 1.f / s;
    for (int j = 0; j < LEN; ++j) Sc[tid][j] *= inv;
  }
  __syncthreads();

  // ---- out = V * attn^T : M=c (2 tiles), N=i (4 tiles), K=j (LEN) ----
  const int ct = wave >> 2;  // 0..1
  const int it = wave & 3;   // 0..3
  v8f acc = {};
#pragma unroll 4
  for (int jj = 0; jj < LEN / 4; ++jj) {
    const int jb = jj * 4 + half * 2;
    v2f a, b;
    a.x = Vlds[ct * 16 + lm][jb];
    a.y = Vlds[ct * 16 + lm][jb + 1];
    b.x = Sc[it * 16 + lm][jb];
    b.y = Sc[it * 16 + lm][jb + 1];
    acc = wmma4(a, b, acc);
  }
#pragma unroll
  for (int r = 0; r < 8; ++r) {
    const int c = ct * 16 + r + half * 8;
    out[(h * C + c) * SL + other * otherStride +
        (i0 + it * 16 + lm) * posStride] = acc[r];
  }
}

// ---------------------------------------------------------------------------
// Channel layernorm over D=128 at each of SL positions: out = LN(xa+xb)
// ---------------------------------------------------------------------------
__global__ __launch_bounds__(256)
void ln_kernel(const float* __restrict__ xa, const float* __restrict__ xb,
               const float* __restrict__ gamma, const float* __restrict__ beta,
               float* __restrict__ out) {
  __shared__ float tbuf[128][64];
  __shared__ float mu_[64];
  __shared__ float rs_[64];

  const int tid = threadIdx.x;
  const int n0  = blockIdx.x * 64;

  for (int e = tid; e < 128 * 64; e += 256) {
    int d = e >> 6, nn = e & 63;
    int g = d * SL + n0 + nn;
    tbuf[d][nn] = xa[g] + xb[g];
  }
  __syncthreads();

  if (tid < 64) {
    float s = 0.f;
    for (int d = 0; d < 128; ++d) s += tbuf[d][tid];
    const float mu = s * (1.f / 128.f);
    float v = 0.f;
    for (int d = 0; d < 128; ++d) {
      float df = tbuf[d][tid] - mu;
      v += df * df;
    }
    mu_[tid] = mu;
    rs_[tid] = rsqrtf(v * (1.f / 128.f) + 1e-5f);
  }
  __syncthreads();

  for (int e = tid; e < 128 * 64; e += 256) {
    int d = e >> 6, nn = e & 63;
    out[d * SL + n0 + nn] =
        (tbuf[d][nn] - mu_[nn]) * rs_[nn] * gamma[d] + beta[d];
  }
}

// ---------------------------------------------------------------------------
extern "C" void kernel_launch(void* const* d_in, const int* in_sizes, int n_in,
                              void* d_out, int out_size, void* d_ws,
                              size_t ws_size, hipStream_t stream) {
  (void)in_sizes; (void)n_in; (void)out_size; (void)ws_size;
  const float* x     = (const float*)d_in[0];
  const float* w_row = (const float*)d_in[1];
  const float* b_row = (const float*)d_in[2];
  const float* w_col = (const float*)d_in[3];
  const float* b_col = (const float*)d_in[4];
  const float* g1    = (const float*)d_in[5];
  const float* be1   = (const float*)d_in[6];
  const float* g2    = (const float*)d_in[7];
  const float* be2   = (const float*)d_in[8];
  float* out = (float*)d_out;

  float* qkv  = (float*)d_ws;          // 3*D*SL floats (48 MB)
  float* tmp  = qkv + 3 * D * SL;      // D*SL floats (16 MB)
  float* out1 = tmp + D * SL;          // D*SL floats (16 MB)

  dim3 gGemm(SL / 64, 3);

  // row attention branch
  qkv_gemm_kernel<<<gGemm, 256, 0, stream>>>(x, w_row, b_row, qkv);
  attn_kernel<256><<<H * S * (L / 64), 256, 0, stream>>>(
      qkv, tmp, /*posStride=*/1, /*otherStride=*/L, /*otherLen=*/S, L / 64);
  ln_kernel<<<SL / 64, 256, 0, stream>>>(x, tmp, g1, be1, out1);

  // column attention branch
  qkv_gemm_kernel<<<gGemm, 256, 0, stream>>>(out1, w_col, b_col, qkv);
  attn_kernel<128><<<H * L * (S / 64), 256, 0, stream>>>(
      qkv, tmp, /*posStride=*/L, /*otherStride=*/1, /*otherLen=*/L, S / 64);
  ln_kernel<<<SL / 64, 256, 0, stream>>>(out1, tmp, g2, be2, out);
}